// SurvPLE_44805098832600
// MI455X (gfx1250) — compile-verified
//
#include <hip/hip_runtime.h>
#include <math.h>

// CDNA5 / gfx1250, wave32. WMMA f32 16x16x4: full-precision matrix path.
typedef float v2f __attribute__((ext_vector_type(2)));
typedef float v8f __attribute__((ext_vector_type(8)));

#define WAVES_PER_BLOCK 8
#define BLOCK_THREADS   256   // 8 waves of 32
#define CHUNK           512   // j-chunk staged in LDS (2 x 2KB)

// ---------------------------------------------------------------- kernel 1
__global__ void cox_exp_kernel(const float* __restrict__ theta,
                               float* __restrict__ w, int n) {
    int i = blockIdx.x * blockDim.x + threadIdx.x;
    if (i < n) w[i] = expf(theta[i]);
}

// ---------------------------------------------------------------- kernel 2
// One wave handles one 16-row tile of the risk-set matvec.
// A[m,k] = (T[j] >= T[i0+m]) ? exp(theta[j]) : 0 ; B = ones(4x16)
// => C[m][*] accumulates S[i0+m] exactly in f32 via v_wmma_f32_16x16x4_f32.
__global__ __launch_bounds__(BLOCK_THREADS)
void cox_tile_kernel(const float* __restrict__ theta,
                     const float* __restrict__ T,
                     const float* __restrict__ ev,
                     const float* __restrict__ w,
                     float* __restrict__ partials,
                     int n) {
    __shared__ __align__(16) float ldsT[CHUNK];
    __shared__ __align__(16) float ldsW[CHUNK];
    __shared__ float wavePart[WAVES_PER_BLOCK];

    const int tid  = threadIdx.x;
    const int lane = tid & 31;
    const int wave = tid >> 5;
    const int m    = lane & 15;          // row within tile (A-matrix M = lane%16)
    const int hi   = (lane >> 4) & 1;    // lane group: K={0,1} vs K={2,3}
    const int k0   = hi << 1;

    const int tile = blockIdx.x * WAVES_PER_BLOCK + wave;
    const int i0   = tile * 16;
    const int irow = i0 + m;
    const float Ti = (irow < n) ? T[irow] : __builtin_inff(); // +inf => row contributes 0

    v8f acc = {0.f, 0.f, 0.f, 0.f, 0.f, 0.f, 0.f, 0.f};
    const v2f ones = {1.0f, 1.0f};

    for (int jc = 0; jc < n; jc += CHUNK) {
        // cooperative stage of T / w chunk into LDS (shared by 8 waves)
        #pragma unroll
        for (int idx = tid; idx < CHUNK; idx += BLOCK_THREADS) {
            int g = jc + idx;
            bool ok = (g < n);
            ldsT[idx] = ok ? T[g] : -__builtin_inff(); // -inf => mask false
            ldsW[idx] = ok ? w[g] : 0.0f;
        }
        __syncthreads();

        #pragma unroll 4
        for (int jj = 0; jj < CHUNK; jj += 4) {
            // lane reads its two K-slots directly (8B-aligned ds_load_b64)
            const v2f tv = *(const v2f*)&ldsT[jj + k0];
            const v2f wv = *(const v2f*)&ldsW[jj + k0];
            v2f a;
            a.x = (tv.x >= Ti) ? wv.x : 0.0f;
            a.y = (tv.y >= Ti) ? wv.y : 0.0f;
            // D = A x B + C   (8 args: neg_a, A, neg_b, B, c_mod, C, reuse_a, reuse_b)
            acc = __builtin_amdgcn_wmma_f32_16x16x4_f32(
                false, a, false, ones, (short)0, acc, false, false);
        }
        __syncthreads();
    }

    // C/D layout (16x16 f32): VGPR r -> M=r on lanes 0-15, M=8+r on lanes 16-31.
    // All 16 columns are identical (B==1), so lane 0 holds rows 0..7 in acc[0..7]
    // and lane 16 holds rows 8..15.  Those two lanes fold in the loss terms.
    float part = 0.0f;
    if (m == 0) {
        const int base = i0 + hi * 8;
        #pragma unroll
        for (int r = 0; r < 8; ++r) {
            int i = base + r;
            if (i < n) {
                float e = ev[i];
                if (e != 0.0f)
                    part += (theta[i] - logf(acc[r])) * e;
            }
        }
    }
    // combine lane 0 + lane 16 within the wave
    part += __shfl(part, 16, 32);
    if (lane == 0) wavePart[wave] = part;
    __syncthreads();

    if (tid == 0) {
        float s = 0.0f;
        #pragma unroll
        for (int v = 0; v < WAVES_PER_BLOCK; ++v) s += wavePart[v];
        partials[blockIdx.x] = s;
    }
}

// ---------------------------------------------------------------- kernel 3
__global__ void cox_reduce_kernel(const float* __restrict__ partials, int np,
                                  float* __restrict__ out, float invN) {
    __shared__ float s[256];
    float v = 0.0f;
    for (int i = threadIdx.x; i < np; i += 256) v += partials[i];
    s[threadIdx.x] = v;
    __syncthreads();
    #pragma unroll
    for (int off = 128; off > 0; off >>= 1) {
        if ((int)threadIdx.x < off) s[threadIdx.x] += s[threadIdx.x + off];
        __syncthreads();
    }
    if (threadIdx.x == 0) out[0] = -s[0] * invN;
}

// ---------------------------------------------------------------- launch
extern "C" void kernel_launch(void* const* d_in, const int* in_sizes, int n_in,
                              void* d_out, int out_size, void* d_ws, size_t ws_size,
                              hipStream_t stream) {
    const float* theta = (const float*)d_in[0]; // risk_scores
    const float* T     = (const float*)d_in[1]; // survival_times
    const float* ev    = (const float*)d_in[2]; // events
    const int n = in_sizes[0];

    float* w        = (float*)d_ws;   // n floats: exp(theta)
    float* partials = w + n;          // one float per tile-block

    const int numTiles = (n + 15) / 16;
    const int grid2    = (numTiles + WAVES_PER_BLOCK - 1) / WAVES_PER_BLOCK;

    cox_exp_kernel<<<(n + 255) / 256, 256, 0, stream>>>(theta, w, n);
    cox_tile_kernel<<<grid2, BLOCK_THREADS, 0, stream>>>(theta, T, ev, w, partials, n);
    cox_reduce_kernel<<<1, 256, 0, stream>>>(partials, grid2, (float*)d_out,
                                             1.0f / (float)n);
}